// HMM_21148418966288
// MI455X (gfx1250) — compile-verified
//
#include <hip/hip_runtime.h>
#include <math.h>

// ---------------- problem constants ----------------
#define NSENS   64
#define OD      128          // obs dim
#define DV      132          // true state dim (4 + 128)
#define DP      144          // padded state dim (multiple of 16)
#define PLD     144          // row stride of P
#define SLD     132          // row stride of S (132 mod 64 = 4 -> conflict-free col walks)
#define ZLD     144          // row stride of Z / W (144 cols: 132 state + r + pad)
#define TSTEPS  4096
#define NTHREADS 512
#define NWAVES   16
#define LOG2PI_F 1.8378770664093454f

typedef float v2f __attribute__((ext_vector_type(2)));
typedef float v8f __attribute__((ext_vector_type(8)));

// NCV process-noise template (scaled by trans_noise^2 at runtime)
__device__ __constant__ float c_Q[16] = {
    1.f/3.f, 0.f,     0.5f, 0.f,
    0.f,     1.f/3.f, 0.f,  0.5f,
    0.5f,    0.f,     1.f,  0.f,
    0.f,     0.5f,    0.f,  1.f };

#define WMMA_F32(a, b, c) \
    __builtin_amdgcn_wmma_f32_16x16x4_f32(false, (a), false, (b), (short)0, (c), false, false)

// ---- WMMA f32 16x16xK tile kernels (wave32; EXEC all ones; operands in LDS) ----
// Layouts per CDNA5 ISA 7.12.2:
//   A 16x4 f32 : lanes 0-15 M=lane {v0:K0,v1:K1}; lanes 16-31 {v0:K2,v1:K3}
//   C/D 16x16  : VGPR v, lanes 0-15 -> M=v, N=lane; lanes 16-31 -> M=v+8

// acc = A(16x16) * B(16x16), row-major; all 8 ds loads issued before the wmma chain
__device__ static inline v8f tile_AB16(const float* A, int lda,
                                       const float* B, int ldb) {
    const int lane = threadIdx.x & 31;
    const int hf = lane >> 4, ln = lane & 15;
    const float* Ar = A + ln * lda + 2 * hf;          // a chunk k: Ar[k], Ar[k+1]
    const float* Bc = B + (2 * hf) * ldb + ln;        // b chunk k: Bc[k*ldb], Bc[(k+1)*ldb]
    v2f a0, a1, a2, a3, b0, b1, b2, b3;
    a0.x = Ar[0];        a0.y = Ar[1];
    a1.x = Ar[4];        a1.y = Ar[5];
    a2.x = Ar[8];        a2.y = Ar[9];
    a3.x = Ar[12];       a3.y = Ar[13];
    b0.x = Bc[0];        b0.y = Bc[ldb];
    b1.x = Bc[4 * ldb];  b1.y = Bc[5 * ldb];
    b2.x = Bc[8 * ldb];  b2.y = Bc[9 * ldb];
    b3.x = Bc[12 * ldb]; b3.y = Bc[13 * ldb];
    v8f c = {0.f,0.f,0.f,0.f,0.f,0.f,0.f,0.f};
    c = WMMA_F32(a0, b0, c);
    c = WMMA_F32(a1, b1, c);
    c = WMMA_F32(a2, b2, c);
    c = WMMA_F32(a3, b3, c);
    return c;
}

// acc = A(16x16) * B(16x16)^T (SYRK-style), row-major
__device__ static inline v8f tile_ABt16(const float* A, int lda,
                                        const float* B, int ldb) {
    const int lane = threadIdx.x & 31;
    const int hf = lane >> 4, ln = lane & 15;
    const float* Ar = A + ln * lda + 2 * hf;
    const float* Br = B + ln * ldb + 2 * hf;          // b chunk k: Br[k], Br[k+1]
    v2f a0, a1, a2, a3, b0, b1, b2, b3;
    a0.x = Ar[0];  a0.y = Ar[1];
    a1.x = Ar[4];  a1.y = Ar[5];
    a2.x = Ar[8];  a2.y = Ar[9];
    a3.x = Ar[12]; a3.y = Ar[13];
    b0.x = Br[0];  b0.y = Br[1];
    b1.x = Br[4];  b1.y = Br[5];
    b2.x = Br[8];  b2.y = Br[9];
    b3.x = Br[12]; b3.y = Br[13];
    v8f c = {0.f,0.f,0.f,0.f,0.f,0.f,0.f,0.f};
    c = WMMA_F32(a0, b0, c);
    c = WMMA_F32(a1, b1, c);
    c = WMMA_F32(a2, b2, c);
    c = WMMA_F32(a3, b3, c);
    return c;
}

// acc = (Z[:,ci:ci+16])^T * Z[:,cj:cj+16] over 128 rows (Z^T Z tile),
// software-pipelined depth 2: next chunk's ds loads overlap current wmma
__device__ static inline v8f tile_ZtZ(const float* Zb, int ldz, int ci, int cj) {
    const int lane = threadIdx.x & 31;
    const int hf = lane >> 4, ln = lane & 15;
    const float* pa = Zb + (2 * hf) * ldz + ci + ln;  // chunk k: pa[k*ldz], pa[(k+1)*ldz]
    const float* pb = Zb + (2 * hf) * ldz + cj + ln;
    v2f a0, b0;
    a0.x = pa[0]; a0.y = pa[ldz];
    b0.x = pb[0]; b0.y = pb[ldz];
    v8f c = {0.f,0.f,0.f,0.f,0.f,0.f,0.f,0.f};
    #pragma unroll 4
    for (int k = 4; k < OD; k += 4) {
        v2f a1, b1;
        a1.x = pa[k * ldz];       a1.y = pa[(k + 1) * ldz];
        b1.x = pb[k * ldz];       b1.y = pb[(k + 1) * ldz];
        c = WMMA_F32(a0, b0, c);
        a0 = a1; b0 = b1;
    }
    c = WMMA_F32(a0, b0, c);
    return c;
}

__global__ __launch_bounds__(NTHREADS, 1)
void kalman_scan_kernel(const float* __restrict__ obs,
                        const float* __restrict__ p_lbs,
                        const float* __restrict__ p_lon,
                        const float* __restrict__ p_ltn,
                        float* __restrict__ out) {
    extern __shared__ float smem[];
    float* P  = smem;                 // DP x PLD  (padded, symmetric, pads = 0)
    float* S  = P  + DP * PLD;        // OD x SLD  (innovation cov -> Cholesky L)
    float* Z  = S  + OD * SLD;        // OD x ZLD  (W then Z = L^-1 W; col 132 = beta)
    float* mv = Z  + OD * ZLD;        // DP        (state mean, padded)
    float* sc = mv + DP;              // [0]=ll_total [1]=sum log diag [2]=beta^T beta

    const int tid  = threadIdx.x;
    // force wave id into an SGPR: tile-ownership checks compile to s_cmp/s_cbranch,
    // keeping EXEC structurally all-ones around every WMMA
    const int wv   = __builtin_amdgcn_readfirstlane(tid >> 5);
    const int lane = tid & 31;
    const int hf   = lane >> 4, ln = lane & 15;

    const float bs2 = expf(2.f * p_lbs[0]);   // bias_scale^2
    const float on2 = expf(2.f * p_lon[0]);   // obs_noise^2
    const float tn2 = expf(2.f * p_ltn[0]);   // trans_noise^2

    // ---- init m0 / P0 ----
    for (int idx = tid; idx < DP * PLD; idx += NTHREADS) {
        int i = idx / PLD, j = idx - i * PLD;
        float v = 0.f;
        if (i == j) v = (i < 4) ? 100.f : (i < DV ? bs2 : 0.f);
        P[idx] = v;
    }
    for (int i = tid; i < DP; i += NTHREADS) mv[i] = 0.f;
    if (tid < 8) sc[tid] = 0.f;
    __syncthreads();

    for (int t = 0; t < TSTEPS; ++t) {
        const float* y = obs + (size_t)t * OD;
        if (tid == 0 && t + 1 < TSTEPS)
            __builtin_prefetch(obs + (size_t)(t + 1) * OD, 0, 3);

        // ================= predict: P = A P A^T + Q, m = A m ===========
        // A[:4,:4] = F^T = [[1,0,1,0],[0,1,0,1],[0,0,1,0],[0,0,0,1]], rest identity
        for (int j = tid; j < DP; j += NTHREADS) {
            P[0 * PLD + j] += P[2 * PLD + j];
            P[1 * PLD + j] += P[3 * PLD + j];
        }
        if (tid == NTHREADS - 1) { mv[0] += mv[2]; mv[1] += mv[3]; }
        __syncthreads();
        for (int i = tid; i < DP; i += NTHREADS) {
            P[i * PLD + 0] += P[i * PLD + 2];
            P[i * PLD + 1] += P[i * PLD + 3];
        }
        __syncthreads();
        if (tid < 16) P[(tid >> 2) * PLD + (tid & 3)] += tn2 * c_Q[tid];
        __syncthreads();

        // ======== W = C P_p  (row m: P[s(m),:] + P[4+m,:]); col 132 = r =========
        for (int idx = tid; idx < OD * ZLD; idx += NTHREADS) {
            int mr = idx / ZLD, j = idx - mr * ZLD;
            int sm = (mr >= NSENS) ? 1 : 0;
            float w;
            if (j < DV)       w = P[sm * PLD + j] + P[(4 + mr) * PLD + j];
            else if (j == DV) w = y[mr] - (mv[sm] + mv[4 + mr]);   // residual
            else              w = 0.f;                              // pad
            Z[mr * ZLD + j] = w;
        }
        __syncthreads();
        // ======== S = C W^T + R : S[m][n] = W[m][s(n)] + W[m][4+n] (+ on2) ======
        for (int idx = tid; idx < OD * OD; idx += NTHREADS) {
            int mr = idx >> 7, n = idx & (OD - 1);
            int sn = (n >= NSENS) ? 1 : 0;
            float s = Z[mr * ZLD + sn] + Z[mr * ZLD + 4 + n];
            if (mr == n) s += on2;
            S[mr * SLD + n] = s;
        }
        __syncthreads();

        // ================= blocked Cholesky S = L L^T (NB=16) =================
        for (int kb = 0; kb < 8; ++kb) {
            float* Lkk = S + (kb * 16) * SLD + kb * 16;
            // -- 16x16 diagonal factor: wave 0, lane-parallel right-looking --
            if (wv == 0) {
                #pragma unroll 1
                for (int j = 0; j < 16; ++j) {
                    float d = sqrtf(Lkk[j * SLD + j]);
                    if (lane == j) Lkk[j * SLD + j] = d;
                    if (lane < 16 && lane > j) {
                        float lij = Lkk[lane * SLD + j] / d;
                        Lkk[lane * SLD + j] = lij;
                        for (int k2 = j + 1; k2 <= lane; ++k2)
                            Lkk[lane * SLD + k2] -= lij * Lkk[k2 * SLD + j];
                    }
                }
            }
            __syncthreads();
            // -- panel: rows below solve X * Lkk^T = S_panel (one thread/row) --
            for (int r = (kb + 1) * 16 + tid; r < OD; r += NTHREADS) {
                float* row = S + r * SLD + kb * 16;
                #pragma unroll 1
                for (int j = 0; j < 16; ++j) {
                    float x = row[j];
                    for (int p = 0; p < j; ++p) x -= row[p] * Lkk[j * SLD + p];
                    row[j] = x / Lkk[j * SLD + j];
                }
            }
            __syncthreads();
            // -- trailing SYRK update via WMMA: S[ib][jb] -= L[ib] L[jb]^T --
            {
                int cnt = 0;
                for (int ib = kb + 1; ib < 8; ++ib)
                    for (int jb = kb + 1; jb <= ib; ++jb, ++cnt)
                        if ((cnt & (NWAVES - 1)) == wv) {
                            v8f acc = tile_ABt16(S + ib * 16 * SLD + kb * 16, SLD,
                                                 S + jb * 16 * SLD + kb * 16, SLD);
                            float* Cb = S + ib * 16 * SLD + jb * 16;
                            #pragma unroll
                            for (int v = 0; v < 8; ++v)
                                Cb[(v + 8 * hf) * SLD + ln] -= acc[v];
                        }
            }
            __syncthreads();
        }
        // log|L| accumulation (ds_add_f32 atomics)
        if (tid < OD) atomicAdd(&sc[1], logf(S[tid * SLD + tid]));

        // ============== blocked forward solve: Z = L^-1 W (in place) ==========
        for (int kb = 0; kb < 8; ++kb) {
            const float* Lkk = S + (kb * 16) * SLD + kb * 16;
            // diagonal 16x16 triangular solve, one thread per column (144 cols)
            for (int cidx = tid; cidx < ZLD; cidx += NTHREADS) {
                float* zc = Z + (kb * 16) * ZLD + cidx;
                float zv[16];
                #pragma unroll
                for (int j = 0; j < 16; ++j) {
                    float x = zc[j * ZLD];
                    #pragma unroll
                    for (int p = 0; p < j; ++p) x -= Lkk[j * SLD + p] * zv[p];
                    zv[j] = x / Lkk[j * SLD + j];
                    zc[j * ZLD] = zv[j];
                }
            }
            __syncthreads();
            // update rows below via WMMA: W[ib] -= L[ib][kb] * Z[kb rows]
            if (kb < 7) {
                int cnt = 0;
                for (int ib = kb + 1; ib < 8; ++ib)
                    for (int jc = 0; jc < 9; ++jc, ++cnt)
                        if ((cnt & (NWAVES - 1)) == wv) {
                            v8f acc = tile_AB16(S + ib * 16 * SLD + kb * 16, SLD,
                                                Z + (kb * 16) * ZLD + jc * 16, ZLD);
                            float* Cb = Z + ib * 16 * ZLD + jc * 16;
                            #pragma unroll
                            for (int v = 0; v < 8; ++v)
                                Cb[(v + 8 * hf) * ZLD + ln] -= acc[v];
                        }
                __syncthreads();
            }
        }

        // ====== fused correction SYRK: G = Z^T Z, lower triangle only =========
        //   P -= G (mirrored); row gm==132 gives m += Z^T beta and beta^T beta
        {
            int cnt = 0;
            for (int I = 0; I < 9; ++I)
                for (int J = 0; J <= I; ++J, ++cnt)
                    if ((cnt & (NWAVES - 1)) == wv) {
                        v8f acc = tile_ZtZ(Z, ZLD, I * 16, J * 16);
                        #pragma unroll
                        for (int v = 0; v < 8; ++v) {
                            int gm = I * 16 + v + 8 * hf;
                            int gn = J * 16 + ln;
                            float g = acc[v];
                            if (gm < DV && gn < DV) {
                                P[gm * PLD + gn] -= g;
                                if (J < I) P[gn * PLD + gm] -= g;   // mirror
                            } else if (gm == DV) {
                                if (gn < DV)       mv[gn] += g;     // K r (symmetry)
                                else if (gn == DV) sc[2] = g;       // beta^T beta
                            }
                        }
                    }
        }
        __syncthreads();
        if (tid == 0) {
            sc[0] += -0.5f * sc[2] - sc[1] - 0.5f * (float)OD * LOG2PI_F;
            sc[1] = 0.f;
        }
        __syncthreads();
    }

    // ---- outputs: [logp, m[0:4], P[0:4,0:4] row-major] = 21 floats ----
    if (tid == 0) out[0] = sc[0];
    if (tid < 4)  out[1 + tid] = mv[tid];
    if (tid < 16) out[5 + tid] = P[(tid >> 2) * PLD + (tid & 3)];
}

extern "C" void kernel_launch(void* const* d_in, const int* in_sizes, int n_in,
                              void* d_out, int out_size, void* d_ws, size_t ws_size,
                              hipStream_t stream) {
    (void)in_sizes; (void)n_in; (void)d_ws; (void)ws_size; (void)out_size;
    const float* obs = (const float*)d_in[0];
    const float* lbs = (const float*)d_in[1];
    const float* lon = (const float*)d_in[2];
    const float* ltn = (const float*)d_in[3];
    float* out = (float*)d_out;

    const size_t smem_bytes =
        (size_t)(DP * PLD + OD * SLD + OD * ZLD + DP + 8) * sizeof(float);

    // single persistent workgroup (16 wave32s on one WGP); whole filter state
    // lives in ~225 KB of the WGP's 320 KB LDS
    kalman_scan_kernel<<<dim3(1), dim3(NTHREADS), smem_bytes, stream>>>(
        obs, lbs, lon, ltn, out);
}